// Quanvolution_69148973465764
// MI455X (gfx1250) — compile-verified
//
#include <hip/hip_runtime.h>
#include <math.h>

typedef float v2f __attribute__((ext_vector_type(2)));
typedef float v8f __attribute__((ext_vector_type(8)));

#define NPATCH   (64 * 128 * 128)   // 1,048,576 patches
#define NGROUPS  (NPATCH / 16)      // 65,536 wave-groups
#define NBLOCKS  (NGROUPS / 8)      // 8 waves per 256-thread block

// ---------------------------------------------------------------------------
// Prologue: build the 16x16 complex circuit unitary U (depends only on the 5
// weights). Thread s simulates the circuit on basis state |s>, producing
// column s of U. s = a*8 + b*4 + c*2 + d  (wire j -> bit mask 8>>j).
// Writes Ur[256] then Ui[256] row-major (U[t*16+s]) into d_ws.
// ---------------------------------------------------------------------------
__global__ void quanv_build_unitary(const float* __restrict__ w,
                                    float* __restrict__ U) {
    int s = threadIdx.x;   // 0..15
    float re[16], im[16];
#pragma unroll
    for (int t = 0; t < 16; ++t) { re[t] = (t == s) ? 1.f : 0.f; im[t] = 0.f; }

    auto rx = [&](float phi, int mask) {   // [[c,-is],[-is,c]]
        float c = __cosf(0.5f * phi), sn = __sinf(0.5f * phi);
#pragma unroll
        for (int i = 0; i < 16; ++i)
            if (!(i & mask)) {
                int j = i | mask;
                float ri = re[i], ii = im[i], rj = re[j], ij = im[j];
                re[i] = c * ri + sn * ij;  im[i] = c * ii - sn * rj;
                re[j] = sn * ii + c * rj;  im[j] = -sn * ri + c * ij;
            }
    };
    auto ry = [&](float phi, int mask) {   // [[c,-s],[s,c]] (real)
        float c = __cosf(0.5f * phi), sn = __sinf(0.5f * phi);
#pragma unroll
        for (int i = 0; i < 16; ++i)
            if (!(i & mask)) {
                int j = i | mask;
                float ri = re[i], ii = im[i], rj = re[j], ij = im[j];
                re[i] = c * ri - sn * rj;  im[i] = c * ii - sn * ij;
                re[j] = sn * ri + c * rj;  im[j] = sn * ii + c * ij;
            }
    };
    auto cnot = [&](int cm, int tm) {      // swap amp within control==1 slice
#pragma unroll
        for (int i = 0; i < 16; ++i)
            if ((i & cm) && !(i & tm)) {
                int j = i | tm;
                float t0 = re[i]; re[i] = re[j]; re[j] = t0;
                float t1 = im[i]; im[i] = im[j]; im[j] = t1;
            }
    };

    rx(w[0], 8);   // RX(w0) wire0
    rx(w[1], 4);   // RX(w1) wire1
    cnot(2, 1);    // CNOT(2,3)
    cnot(8, 2);    // CNOT(0,2)
    cnot(8, 1);    // CNOT(0,3)
    ry(w[3], 8);   // RY(w3) wire0
    ry(w[4], 1);   // RY(w4) wire3

#pragma unroll
    for (int t = 0; t < 16; ++t) {
        U[t * 16 + s]       = re[t];   // Ur row-major
        U[256 + t * 16 + s] = im[t];   // Ui row-major
    }
}

// ---------------------------------------------------------------------------
// ds_swizzle_b32, group-of-32 mode: new_lane = lane ^ M  (and=0x1f, or=0)
// ---------------------------------------------------------------------------
template <int M>
__device__ __forceinline__ float swz_xor(float v) {
    int r = __builtin_amdgcn_ds_swizzle(__float_as_int(v), (M << 10) | 0x1f);
    return __int_as_float(r);
}

// ---------------------------------------------------------------------------
// Main kernel: one wave = 16 patches.
//   ampT[t,p] = sum_s U[t,s] * state0[p,s]   via 8x V_WMMA_F32_16X16X4_F32
//               (A = U fragment from LDS, B = encoded-state fragment)
//   probs     = Re^2 + Im^2   (lane owns patch p=lane%16, states t=i+8*hiH)
//   z_j       = partial in-register FWHT (u=0,1,2,4) + one xor-16 exchange
// ---------------------------------------------------------------------------
__global__ __launch_bounds__(256) void quanv_main(const float* __restrict__ x,
                                                  const float* __restrict__ Ubuf,
                                                  float* __restrict__ out) {
    __shared__ __align__(16) float sU[512];      // Ur[256] | Ui[256]

    int tid = threadIdx.x;
    sU[tid]       = Ubuf[tid];
    sU[tid + 256] = Ubuf[tid + 256];
    __syncthreads();

    int wave = tid >> 5;
    int lane = tid & 31;
    int row  = lane & 15;          // patch col (N) == U row t (M)
    int hiH  = lane >> 4;          // 0 lower half, 1 upper half

    int group = blockIdx.x * 8 + wave;    // 16-patch group id
    int p     = group * 16 + row;         // global patch index
    int b     = p >> 14;                  // image
    int rem   = p & 16383;
    int pj    = rem >> 7;                 // patch row in image
    int pk    = rem & 127;                // patch col in image

    // ---- load 2x2 patch (two float2 loads, 8B aligned, coalesced) ----
    const float2* r0 = (const float2*)(x + ((size_t)(b * 256 + 2 * pj)) * 256 + 2 * pk);
    const float2* r1 = (const float2*)(x + ((size_t)(b * 256 + 2 * pj + 1)) * 256 + 2 * pk);
    float2 t0 = r0[0];
    float2 t1 = r1[0];

    // ---- RY(pi*x) encoding: wire j amplitudes (cos(pi x/2), sin(pi x/2)) ----
    float c0, s0, c1, s1, c2, s2, c3, s3;
    __sincosf(1.57079632679489662f * t0.x, &s0, &c0);
    __sincosf(1.57079632679489662f * t0.y, &s1, &c1);
    __sincosf(1.57079632679489662f * t1.x, &s2, &c2);
    __sincosf(1.57079632679489662f * t1.y, &s3, &c3);

    // Upper-half lanes hold the K+2 slice of each chunk, i.e. states with
    // wire-2 bit c == 1.  Select the e2 factor once per lane (1 cndmask),
    // then every B-fragment element has a compile-time index.
    float g2 = hiH ? s2 : c2;
    float gB[4] = {(c0 * c1) * g2,    // q=0: (a,b)=(0,0)
                   (c0 * s1) * g2,    // q=1: (a,b)=(0,1)
                   (s0 * c1) * g2,    // q=2: (a,b)=(1,0)
                   (s0 * s1) * g2};   // q=3: (a,b)=(1,1)

    // ---- ampT = U @ state0^T : 4 K-chunks x (Re, Im) ----
    const float2* sU2  = (const float2*)sU;
    int           aIdx = row * 8 + hiH;   // float2 index of U fragment, chunk 0
    v8f cRe = {0.f, 0.f, 0.f, 0.f, 0.f, 0.f, 0.f, 0.f};
    v8f cIm = {0.f, 0.f, 0.f, 0.f, 0.f, 0.f, 0.f, 0.f};
#pragma unroll
    for (int q = 0; q < 4; ++q) {
        v2f bst;
        bst.x = gB[q] * c3;               // d = 0
        bst.y = gB[q] * s3;               // d = 1
        float2 arf = sU2[aIdx + 2 * q];          // Ur[t, 4q+kh .. +1]
        float2 aif = sU2[128 + aIdx + 2 * q];    // Ui[t, 4q+kh .. +1]
        v2f ar; ar.x = arf.x; ar.y = arf.y;
        v2f ai; ai.x = aif.x; ai.y = aif.y;
        cRe = __builtin_amdgcn_wmma_f32_16x16x4_f32(false, ar, false, bst,
                                                    (short)0, cRe, false, false);
        cIm = __builtin_amdgcn_wmma_f32_16x16x4_f32(false, ai, false, bst,
                                                    (short)0, cIm, false, false);
    }

    // ---- probs: lane owns patch row, states t = i + 8*hiH ----
    float pr[8];
#pragma unroll
    for (int i = 0; i < 8; ++i)
        pr[i] = cRe[i] * cRe[i] + cIm[i] * cIm[i];

    // ---- partial FWHT over local 3 bits of t (need u = 0,1,2,4 only) ----
    float a01 = pr[0] + pr[1], d01 = pr[0] - pr[1];
    float a23 = pr[2] + pr[3], d23 = pr[2] - pr[3];
    float a45 = pr[4] + pr[5], d45 = pr[4] - pr[5];
    float a67 = pr[6] + pr[7], d67 = pr[6] - pr[7];
    float A = a01 + a23, B = a45 + a67;
    float C = d01 + d23, D = d45 + d67;
    float E = a01 - a23, F = a45 - a67;
    float T0 = A + B;   // plain sum
    float T4 = A - B;   // sign by t-bit2
    float T2 = E + F;   // sign by t-bit1
    float T1 = C + D;   // sign by t-bit0

    // ---- exchange with partner half (t-bit3) and combine ----
    float pT0 = swz_xor<16>(T0);
    float pT4 = swz_xor<16>(T4);
    float pT2 = swz_xor<16>(T2);
    float pT1 = swz_xor<16>(T1);
    float sg  = hiH ? -1.f : 1.f;
    float z0  = sg * (T0 - pT0);   // wire0 <-> t-bit3
    float z1  = T4 + pT4;          // wire1 <-> t-bit2
    float z2  = T2 + pT2;          // wire2 <-> t-bit1
    float z3  = T1 + pT1;          // wire3 <-> t-bit0

    // ---- coalesced float4 store: one patch (4 floats) per lane 0..15 ----
    if (lane < 16) {
        float4 v = make_float4(z0, z1, z2, z3);
        ((float4*)out)[group * 16 + row] = v;
    }
}

extern "C" void kernel_launch(void* const* d_in, const int* in_sizes, int n_in,
                              void* d_out, int out_size, void* d_ws, size_t ws_size,
                              hipStream_t stream) {
    const float* x = (const float*)d_in[0];   // (64,1,256,256) f32
    const float* w = (const float*)d_in[1];   // (1,5) f32
    float* U   = (float*)d_ws;                // 512 floats: Ur | Ui
    float* out = (float*)d_out;               // (64,128,128,4) f32

    quanv_build_unitary<<<dim3(1), dim3(16), 0, stream>>>(w, U);
    quanv_main<<<dim3(NBLOCKS), dim3(256), 0, stream>>>(x, U, out);
}